// NTfm3D_91130616086852
// MI455X (gfx1250) — compile-verified
//
#include <hip/hip_runtime.h>

typedef __attribute__((ext_vector_type(2))) float v2f;
typedef __attribute__((ext_vector_type(8))) float v8f;

#define HW    (480 * 640)   // 307200 pixels per batch image
#define KMSK  16
#define WAVES 8
#define ITERS 4
#define PITCH 20            // dwords per LDS row: 80B -> 16B aligned, bank-conflict-free
#define TILE_DW (16 * PITCH)

// out[b,c,p] = sum_k masks[b,k,p] * ( R[b,k,c,:] . points[b,:,p] + t[b,k,c] )
// Reformulated: D = Masks(16pix x 16k) x T(16k x 12)  via v_wmma_f32_16x16x4_f32,
// then out_c = D[p, 4c..4c+2] . (px,py,pz) + D[p, 4c+3].
__global__ __launch_bounds__(256) void ntfm3d_wmma_kernel(
    const float* __restrict__ points,      // (B,3,H,W)
    const float* __restrict__ masks,       // (B,16,H,W)
    const float* __restrict__ transforms,  // (B,16,3,4)
    float* __restrict__ out)               // (B,3,H,W)
{
    __shared__ __align__(16) float lds[WAVES * 2 * TILE_DW];

    const int lane = threadIdx.x & 31;
    const int wave = threadIdx.x >> 5;
    const int b    = blockIdx.y;
    const int half = lane >> 4;    // 0: lanes 0-15, 1: lanes 16-31
    const int l15  = lane & 15;

    const float* maskB = masks      + (size_t)b * KMSK * HW;
    const float* trB   = transforms + (size_t)b * KMSK * 12;
    const float* ptB   = points     + (size_t)b * 3 * HW;
    float*       outB  = out        + (size_t)b * 3 * HW;

    // ---- B operand: T rows, loaded once per wave, reused for all pixels ----
    // B 4x16 layout: VGPR0 = rows k0 (lanes 0-15) / k0+2 (lanes 16-31),
    //                VGPR1 = rows k0+1 / k0+3.  Cols 12..15 zero-padded.
    v2f bmat[4];
    {
        const bool valid = (l15 < 12);
        #pragma unroll
        for (int kk = 0; kk < 4; ++kk) {
            const int k0 = kk * 4 + half * 2;
            bmat[kk].x = valid ? trB[(size_t)k0       * 12 + l15] : 0.0f;
            bmat[kk].y = valid ? trB[(size_t)(k0 + 1) * 12 + l15] : 0.0f;
        }
    }

    float* myLds = lds + wave * 2 * TILE_DW;
    const int waveBasePix = blockIdx.x * (WAVES * ITERS * 32) + wave * (ITERS * 32);

    for (int it = 0; it < ITERS; ++it) {
        const int p0 = waveBasePix + it * 32;   // 32 pixels this iteration

        v8f acc0 = {0.f, 0.f, 0.f, 0.f, 0.f, 0.f, 0.f, 0.f};
        v8f acc1 = {0.f, 0.f, 0.f, 0.f, 0.f, 0.f, 0.f, 0.f};

        // ---- A operand: mask tiles (16 pixels x 4 k) per step, coalesced ----
        #pragma unroll
        for (int kk = 0; kk < 4; ++kk) {
            const int   krow = kk * 4 + half * 2;
            const float* mp  = maskB + (size_t)krow * HW;
            v2f a0, a1;
            a0.x = mp[p0 + l15];            // K-local 0 (or 2), rows = pixels p0..p0+15
            a0.y = mp[HW + p0 + l15];       // K-local 1 (or 3)
            a1.x = mp[p0 + 16 + l15];       // second tile: pixels p0+16..p0+31
            a1.y = mp[HW + p0 + 16 + l15];

            acc0 = __builtin_amdgcn_wmma_f32_16x16x4_f32(
                       false, a0, false, bmat[kk], (short)0, acc0, false, false);
            acc1 = __builtin_amdgcn_wmma_f32_16x16x4_f32(
                       false, a1, false, bmat[kk], (short)0, acc1, false, false);
        }

        // ---- Transpose D tiles through LDS (rows striped across lanes) ----
        // acc[v] at (half,l15) is D[row = v + half*8, col = l15].
        {
            float* t0 = myLds + (half * 8) * PITCH + l15;
            float* t1 = t0 + TILE_DW;
            #pragma unroll
            for (int v = 0; v < 8; ++v) {
                t0[v * PITCH] = acc0[v];
                t1[v * PITCH] = acc1[v];
            }
        }
        asm volatile("s_wait_dscnt 0" ::: "memory");

        // ---- Epilogue: lane owns pixel p0+lane; 3x (4-vector dot) ----
        const float* row = myLds + half * TILE_DW + l15 * PITCH;
        const float4 q0 = *(const float4*)(row);       // c=0: R00 R01 R02 t0
        const float4 q1 = *(const float4*)(row + 4);   // c=1
        const float4 q2 = *(const float4*)(row + 8);   // c=2

        const int pix = p0 + lane;
        const float px = ptB[pix];
        const float py = ptB[HW + pix];
        const float pz = ptB[2 * HW + pix];

        outB[pix]          = q0.x * px + q0.y * py + q0.z * pz + q0.w;
        outB[HW + pix]     = q1.x * px + q1.y * py + q1.z * pz + q1.w;
        outB[2 * HW + pix] = q2.x * px + q2.y * py + q2.z * pz + q2.w;

        // Drain LDS reads before next iteration overwrites the tiles.
        asm volatile("s_wait_dscnt 0" ::: "memory");
    }
}

extern "C" void kernel_launch(void* const* d_in, const int* in_sizes, int n_in,
                              void* d_out, int out_size, void* d_ws, size_t ws_size,
                              hipStream_t stream) {
    (void)in_sizes; (void)n_in; (void)out_size; (void)d_ws; (void)ws_size;
    const float* points     = (const float*)d_in[0];
    const float* masks      = (const float*)d_in[1];
    const float* transforms = (const float*)d_in[2];
    float*       out        = (float*)d_out;

    // 307200 pixels / (8 waves * 4 iters * 32 pixels) = 300 blocks per batch
    dim3 grid(HW / (WAVES * ITERS * 32), 16);
    dim3 block(256);
    hipLaunchKernelGGL(ntfm3d_wmma_kernel, grid, block, 0, stream,
                       points, masks, transforms, out);
}